// PaiNN_42365557408035
// MI455X (gfx1250) — compile-verified
//
#include <hip/hip_runtime.h>
#include <math.h>

typedef __attribute__((ext_vector_type(2))) float v2f;
typedef __attribute__((ext_vector_type(8))) float v8f;

#define F_DIM 128
#define F3    384
#define NRBF  20
#define CUTOFF_F 5.0f
#define PI_F 3.14159265358979323846f

// ---------------------------------------------------------------------------
// Kernel 0: initialize outputs. out_emb = emb_table[z], out_eq = eq (input).
// ---------------------------------------------------------------------------
__global__ void painn_init(const float* __restrict__ emb_table,
                           const int* __restrict__ z,
                           const float* __restrict__ eq_in,
                           float* __restrict__ out_emb,
                           float* __restrict__ out_eq,
                           int n_nodes)
{
    size_t i = (size_t)blockIdx.x * blockDim.x + threadIdx.x;
    size_t total_emb = (size_t)n_nodes * F_DIM;
    size_t total_eq  = (size_t)n_nodes * F3;     // 128*3 per node
    if (i < total_emb) {
        int node = (int)(i >> 7);
        int f    = (int)(i & 127);
        out_emb[i] = emb_table[(size_t)z[node] * F_DIM + f];
    } else if (i < total_emb + total_eq) {
        size_t j = i - total_emb;
        out_eq[j] = eq_in[j];
    }
}

// ---------------------------------------------------------------------------
// Kernel 1: per-node  phi = silu(emb_table[z] @ W1 + b1) @ W2 + b2  [N, 384]
// One 128-thread block (4 waves) per 16 node rows. fp32 WMMA 16x16x4.
// ---------------------------------------------------------------------------
__global__ __launch_bounds__(128) void painn_node_phi(
    const float* __restrict__ emb_table, const int* __restrict__ z,
    const float* __restrict__ w1, const float* __restrict__ b1,
    const float* __restrict__ w2, const float* __restrict__ b2,
    float* __restrict__ phi_out, int n_nodes)
{
    __shared__ float Xs[16 * F_DIM];   // gathered embeddings, row-major [16][128]
    __shared__ float Hs[16 * F_DIM];   // silu(X@W1+b1)

    const int tid  = threadIdx.x;
    const int wave = tid >> 5;
    const int lane = tid & 31;
    const int half = lane >> 4;        // 0: lanes 0-15, 1: lanes 16-31
    const int m    = lane & 15;
    const int row0 = blockIdx.x * 16;

    // Stage gathered X tile: thread 'tid' loads column tid of all 16 rows.
    for (int i = 0; i < 16; ++i) {
        int r  = row0 + i;
        int zr = z[r < n_nodes ? r : (n_nodes - 1)];
        Xs[i * F_DIM + tid] = emb_table[(size_t)zr * F_DIM + tid];
    }
    __syncthreads();

    // ---- GEMM1: H[16,128] = X @ W1, each wave owns 2 column tiles ----------
    for (int ct = wave * 2; ct < wave * 2 + 2; ++ct) {
        const int colbase = ct * 16;
        v8f acc = {};
        #pragma unroll
        for (int k0 = 0; k0 < F_DIM; k0 += 4) {
            const int ka = k0 + 2 * half;           // K pair owned by this half
            v2f a, b;
            a.x = Xs[m * F_DIM + ka];               // A: (M=m, K=ka)
            a.y = Xs[m * F_DIM + ka + 1];           //    (M=m, K=ka+1)
            b.x = w1[(size_t)ka * F_DIM + colbase + m];        // B: (K=ka,   N=m)
            b.y = w1[(size_t)(ka + 1) * F_DIM + colbase + m];  //    (K=ka+1, N=m)
            acc = __builtin_amdgcn_wmma_f32_16x16x4_f32(
                      false, a, false, b, (short)0, acc, false, false);
        }
        const int col  = colbase + m;
        const float bias = b1[col];
        #pragma unroll
        for (int v = 0; v < 8; ++v) {
            const int row = v + 8 * half;           // C vgpr v -> M = v + 8*half
            float x = acc[v] + bias;
            float s = x / (1.0f + __expf(-x));      // silu
            Hs[row * F_DIM + col] = s;
        }
    }
    __syncthreads();

    // ---- GEMM2: phi[16,384] = H @ W2, each wave owns 6 of 24 column tiles --
    for (int ct = wave; ct < 24; ct += 4) {
        const int colbase = ct * 16;
        v8f acc = {};
        #pragma unroll
        for (int k0 = 0; k0 < F_DIM; k0 += 4) {
            const int ka = k0 + 2 * half;
            v2f a, b;
            a.x = Hs[m * F_DIM + ka];
            a.y = Hs[m * F_DIM + ka + 1];
            b.x = w2[(size_t)ka * F3 + colbase + m];
            b.y = w2[(size_t)(ka + 1) * F3 + colbase + m];
            acc = __builtin_amdgcn_wmma_f32_16x16x4_f32(
                      false, a, false, b, (short)0, acc, false, false);
        }
        const int col  = colbase + m;
        const float bias = b2[col];
        #pragma unroll
        for (int v = 0; v < 8; ++v) {
            const int row = row0 + v + 8 * half;
            if (row < n_nodes)
                phi_out[(size_t)row * F3 + col] = acc[v] + bias;
        }
    }
}

// ---------------------------------------------------------------------------
// Kernel 2: edge pass. One wave per 4 edges; lane owns columns c = lane+32k.
// ---------------------------------------------------------------------------
__global__ __launch_bounds__(256) void painn_edge(
    const float* __restrict__ pos, const float* __restrict__ eq_in,
    const float* __restrict__ phi,
    const float* __restrict__ w_rbf, const float* __restrict__ b_rbf,
    const int* __restrict__ esrc, const int* __restrict__ edst,
    float* __restrict__ out_emb, float* __restrict__ out_eq,
    int n_edges)
{
    __shared__ float wr[NRBF * F3];   // 30720 B
    __shared__ float br[F3];

    const int tid = threadIdx.x;
    for (int i = tid; i < NRBF * F3; i += 256) wr[i] = w_rbf[i];
    for (int i = tid; i < F3; i += 256)        br[i] = b_rbf[i];
    __syncthreads();

    const int wave = tid >> 5;
    const int lane = tid & 31;
    const int e0   = blockIdx.x * 32 + wave * 4;

    for (int ei = 0; ei < 4; ++ei) {
        const int e = e0 + ei;
        if (e >= n_edges) break;                 // wave-uniform
        const int s = esrc[e];
        const int d = edst[e];

        const float rx = pos[(size_t)d * 3 + 0] - pos[(size_t)s * 3 + 0];
        const float ry = pos[(size_t)d * 3 + 1] - pos[(size_t)s * 3 + 1];
        const float rz = pos[(size_t)d * 3 + 2] - pos[(size_t)s * 3 + 2];
        const float dist  = sqrtf(rx * rx + ry * ry + rz * rz);
        const float inv_d = 1.0f / dist;
        const float t     = PI_F * dist / CUTOFF_F;

        float rbf[NRBF];
        #pragma unroll
        for (int r = 0; r < NRBF; ++r)
            rbf[r] = __sinf((float)(r + 1) * t) * inv_d;

        const float* phis = phi + (size_t)s * F3;
        float split[12];
        #pragma unroll
        for (int k = 0; k < 12; ++k) {
            const int c = lane + 32 * k;
            float w = br[c];
            #pragma unroll
            for (int r = 0; r < NRBF; ++r) w += rbf[r] * wr[r * F3 + c];
            split[k] = phis[c] * w;
        }

        const float rdx = rx * dist, rdy = ry * dist, rdz = rz * dist;
        #pragma unroll
        for (int k = 0; k < 4; ++k) {
            const int f = lane + 32 * k;
            atomicAdd(&out_emb[(size_t)d * F_DIM + f], split[k]);
            const size_t ib = ((size_t)s * F_DIM + f) * 3;
            const size_t ob = ((size_t)d * F_DIM + f) * 3;
            const float sm = split[k + 4];   // gating of eq[src]
            const float sh = split[k + 8];   // gating of rel*dist
            atomicAdd(&out_eq[ob + 0], eq_in[ib + 0] * sm + sh * rdx);
            atomicAdd(&out_eq[ob + 1], eq_in[ib + 1] * sm + sh * rdy);
            atomicAdd(&out_eq[ob + 2], eq_in[ib + 2] * sm + sh * rdz);
        }
    }
}

// ---------------------------------------------------------------------------
extern "C" void kernel_launch(void* const* d_in, const int* in_sizes, int n_in,
                              void* d_out, int out_size, void* d_ws, size_t ws_size,
                              hipStream_t stream)
{
    const float* pos       = (const float*)d_in[0];
    const float* eq        = (const float*)d_in[1];
    const float* emb_table = (const float*)d_in[2];
    const float* w1        = (const float*)d_in[3];
    const float* b1        = (const float*)d_in[4];
    const float* w2        = (const float*)d_in[5];
    const float* b2        = (const float*)d_in[6];
    const float* w_rbf     = (const float*)d_in[7];
    const float* b_rbf     = (const float*)d_in[8];
    const int*   z         = (const int*)d_in[9];
    const int*   esrc      = (const int*)d_in[10];
    const int*   edst      = (const int*)d_in[11];

    const int n_nodes = in_sizes[0] / 3;
    const int n_edges = in_sizes[10];

    float* out_emb = (float*)d_out;
    float* out_eq  = out_emb + (size_t)n_nodes * F_DIM;
    float* phi_ws  = (float*)d_ws;             // n_nodes * 384 floats (76.8 MB)

    // 1) init outputs (emb + eq passthrough)
    const size_t tot = (size_t)n_nodes * (F_DIM + F3);
    painn_init<<<(unsigned)((tot + 255) / 256), 256, 0, stream>>>(
        emb_table, z, eq, out_emb, out_eq, n_nodes);

    // 2) per-node phi via fp32 WMMA GEMMs
    painn_node_phi<<<(n_nodes + 15) / 16, 128, 0, stream>>>(
        emb_table, z, w1, b1, w2, b2, phi_ws, n_nodes);

    // 3) edge message pass + scatter-add
    painn_edge<<<(n_edges + 31) / 32, 256, 0, stream>>>(
        pos, eq, phi_ws, w_rbf, b_rbf, esrc, edst, out_emb, out_eq, n_edges);
}